// ScaleAwareROIPooling_6201932775780
// MI455X (gfx1250) — compile-verified
//
#include <hip/hip_runtime.h>

// Scale-aware ROI pooling for MI455X (gfx1250).
// Per (box, batch) workgroup: stage selected-level region into LDS [ch][K]
// (coalesced global row loads, conflict-free stride-97 LDS), then reduce with
// fp32 WMMA (V_WMMA_F32_16X16X4_F32): A = 16ch x 4pix tile, B = weight vector
// broadcast across all 16 N columns, accumulate over K in steps of 4.
// K loop unrolled x2 so one s_wait_dscnt covers two WMMA k-steps.

#define KMAX 96   // K = region pixels; analysis bound is ~64, 96 is safe margin
#define KSTR 97   // LDS pad: 97 mod 64 == 33 (odd) -> conflict-free strided access

typedef __attribute__((ext_vector_type(2))) float v2f;
typedef __attribute__((ext_vector_type(8))) float v8f;

#if defined(__has_builtin)
#if __has_builtin(__builtin_amdgcn_wmma_f32_16x16x4_f32)
#define USE_WMMA_F32 1
#endif
#endif

__global__ __launch_bounds__(256) void roi_pool_kernel(
    const float* __restrict__ p3, const float* __restrict__ p4,
    const float* __restrict__ p5, const float* __restrict__ boxes,
    float* __restrict__ out, int* __restrict__ scale_out)
{
  __shared__ float lds_f[256][KSTR];  // [channel][pixel k], ~99.3 KB
  __shared__ float s_wy[40];
  __shared__ float s_wx[40];
  __shared__ float s_wp[KMAX];        // combined per-pixel weight (incl. 1/49)

  const int b    = blockIdx.x;   // batch 0..15
  const int box  = blockIdx.y;   // 0..999
  const int tid  = threadIdx.x;
  const int lane = tid & 31;
  const int wave = tid >> 5;

  // ---- box geometry (uniform across block; matches reference exactly) ----
  const float x1 = boxes[box * 4 + 0];
  const float y1 = boxes[box * 4 + 1];
  const float x2 = boxes[box * 4 + 2];
  const float y2 = boxes[box * 4 + 3];
  const float area = (x2 - x1) * (y2 - y1);
  const int level = (area < 1024.0f) ? 0 : ((area < 9216.0f) ? 1 : 2);
  const float* fm = (level == 0) ? p3 : ((level == 1) ? p4 : p5);
  const int W = 80 >> level;
  const int H = W;

  int fx1 = (int)floorf(x1 / 640.0f * (float)W);
  int fy1 = (int)floorf(y1 / 640.0f * (float)H);
  fx1 = min(max(fx1, 0), W - 1);
  fy1 = min(max(fy1, 0), H - 1);
  int fx2 = (int)floorf(x2 / 640.0f * (float)W);
  int fy2 = (int)floorf(y2 / 640.0f * (float)H);
  fx2 = max(fx1 + 1, min(fx2, W));
  fy2 = max(fy1 + 1, min(fy2, H));
  int       h = fy2 - fy1;          // region rows  (<= 32)
  const int w = fx2 - fx1;          // region cols  (<= 32)
  if (h * w > KMAX) h = KMAX / w;   // defensive clamp (unreachable per bounds)
  const int K   = w * h;            // region pixels (<= ~64 by area bounds)
  const int KR8 = (K + 7) & ~7;     // K rounded up to 2 WMMA k-steps (<= 96)

  // ---- separable adaptive-pool weights ----
  if (tid < h) {
    const int r = tid;
    float acc = 0.0f;
    #pragma unroll
    for (int i = 0; i < 7; i++) {
      const int sy = i * h / 7;
      const int ey = ((i + 1) * h + 6) / 7;
      if (r >= sy && r < ey) acc += 1.0f / (float)(ey - sy);
    }
    s_wy[r] = acc;
  }
  if (tid >= 64 && (tid - 64) < w) {
    const int c = tid - 64;
    float acc = 0.0f;
    #pragma unroll
    for (int i = 0; i < 7; i++) {
      const int sx = i * w / 7;
      const int ex = ((i + 1) * w + 6) / 7;
      if (c >= sx && c < ex) acc += 1.0f / (float)(ex - sx);
    }
    s_wx[c] = acc;
  }
  __syncthreads();

  // combined per-pixel weight, zero-padded to KR8
  for (int k = tid; k < KR8; k += 256) {
    float v = 0.0f;
    if (k < K) v = s_wy[k / w] * s_wx[k % w] * (1.0f / 49.0f);
    s_wp[k] = v;
  }

  // ---- stage region into LDS: coalesced row loads, [ch][k] layout ----
  const size_t HW = (size_t)H * W;
  const float* fmb = fm + (size_t)(b * 256) * HW + (size_t)fy1 * W + fx1;
  const int pad = KR8 - K;          // 0..7, NaN-safe zero pad
  for (int ch = wave; ch < 256; ch += 8) {
    const float* rowp = fmb + (size_t)ch * HW;
    for (int ry = 0; ry < h; ry++) {
      if (lane < w) lds_f[ch][ry * w + lane] = rowp[(size_t)ry * W + lane];
    }
    if (lane < pad) lds_f[ch][K + lane] = 0.0f;
  }
  __syncthreads();

  float* outp = out + ((size_t)box * 16 + b) * 256;

#if defined(USE_WMMA_F32)
  // ---- fp32 WMMA GEMV: D(16x16) = A(16ch x 4pix) * B(4pix x 16), all N
  // columns carry the same weight vector, so every column of D equals out[ch].
  // A layout (ISA 7.12.2): lanes 0-15 hold M=0..15 {K=k,k+1}; lanes 16-31
  // hold M=0..15 {K=k+2,k+3}. B mirrored on K (rows 0,2 | 1,3).
  const int half = lane >> 4;           // 0: K pair (k,k+1), 1: (k+2,k+3)
  const int m    = lane & 15;
  for (int t = wave; t < 16; t += 8) {  // 16 channel tiles, 8 waves
    v8f acc = {0.f, 0.f, 0.f, 0.f, 0.f, 0.f, 0.f, 0.f};
    const float* chp = &lds_f[t * 16 + m][0];
    for (int k = 0; k < KR8; k += 8) {
      const int kb0 = k + 2 * half;
      const int kb1 = kb0 + 4;
      // issue all four LDS reads before the WMMA pair -> single dscnt wait
      v2f a0; a0.x = chp[kb0];   a0.y = chp[kb0 + 1];
      v2f b0; b0.x = s_wp[kb0];  b0.y = s_wp[kb0 + 1];
      v2f a1; a1.x = chp[kb1];   a1.y = chp[kb1 + 1];
      v2f b1; b1.x = s_wp[kb1];  b1.y = s_wp[kb1 + 1];
      acc = __builtin_amdgcn_wmma_f32_16x16x4_f32(
          false, a0, false, b0, (short)0, acc, false, false);
      acc = __builtin_amdgcn_wmma_f32_16x16x4_f32(
          false, a1, false, b1, (short)0, acc, false, false);
    }
    // D layout: VGPR j -> M=j (lanes 0-15) / M=8+j (lanes 16-31); cols equal.
    if (m == 0) {
      float* o = outp + t * 16 + half * 8;
      #pragma unroll
      for (int j = 0; j < 8; j++) o[j] = acc[j];
    }
  }
#else
  // VALU fallback: one channel per thread
  {
    float acc = 0.0f;
    for (int k = 0; k < K; k++) acc += lds_f[tid][k] * s_wp[k];
    outp[tid] = acc;
  }
#endif

  if (b == 0 && tid == 0) scale_out[box] = level;
}

extern "C" void kernel_launch(void* const* d_in, const int* in_sizes, int n_in,
                              void* d_out, int out_size, void* d_ws, size_t ws_size,
                              hipStream_t stream) {
  (void)in_sizes; (void)n_in; (void)out_size; (void)d_ws; (void)ws_size;
  const float* p3    = (const float*)d_in[0];
  const float* p4    = (const float*)d_in[1];
  const float* p5    = (const float*)d_in[2];
  const float* boxes = (const float*)d_in[3];

  float* feats = (float*)d_out;                         // [1000][16][256]
  int*   sid   = (int*)d_out + (size_t)1000 * 16 * 256; // [1000] int32 bits

  dim3 grid(16, 1000);
  roi_pool_kernel<<<grid, 256, 0, stream>>>(p3, p4, p5, boxes, feats, sid);
}